// GatedGraphConv_23235773071822
// MI455X (gfx1250) — compile-verified
//
#include <hip/hip_runtime.h>
#include <hip/hip_bf16.h>

// ---------------------------------------------------------------------------
// GatedGraphConv for MI455X (gfx1250, wave32, WMMA)
//   update = segment_sum(w_e * x[src] -> (dst, etype))        [N, 256]
//   hidden = relu(update @ mlp_W^T + mlp_b)                   [N, 256]  (WMMA f32)
//   gi = hidden @ w_ih^T + b_ih ; gh = x @ w_hh^T + b_hh      [N, 192]  (WMMA f32, fused)
//   out = GRU gates (sigmoid/tanh elementwise)                [N, 64]
// fp32 WMMA (16x16x4) keeps full reference precision; the op is HBM/L2-atomic
// bound anyway (25 GFLOP vs ~0.5 GB of traffic at 23.3 TB/s).
// ---------------------------------------------------------------------------

#define N_NODES 100000
#define N_EDGES 1000000
#define DIM     64      // input dim / GRU hidden size
#define HID     256     // mlp hidden = NUM_EDGE_TYPE * DIM
#define G3      192     // 3 * DIM gate width
#define NTYPE   4

typedef __attribute__((ext_vector_type(2))) float v2f;
typedef __attribute__((ext_vector_type(8))) float v8f;

// --------------------------- zero the scatter buffer -----------------------
__global__ __launch_bounds__(256) void zero_kernel(float4* __restrict__ p, int n4) {
    int i = blockIdx.x * 256 + threadIdx.x;
    if (i < n4) p[i] = make_float4(0.f, 0.f, 0.f, 0.f);
}

// --------------------------- weight transpose ------------------------------
// dst[c * rows + r] = src[r * cols + c]   (src is [rows, cols])
__global__ __launch_bounds__(256) void transpose_kernel(const float* __restrict__ src,
                                                        float* __restrict__ dst,
                                                        int rows, int cols) {
    int i = blockIdx.x * 256 + threadIdx.x;
    if (i < rows * cols) {
        int r = i / cols, c = i % cols;
        dst[c * rows + r] = src[i];
    }
}

// --------------------------- edge scatter-add ------------------------------
// One wave per edge: 32 lanes x float2 = 64 features. unsafeAtomicAdd lowers
// to global_atomic_add_f32 (L2 atomic units; update[] ~fits in 192MB L2).
__global__ __launch_bounds__(256) void scatter_kernel(const float* __restrict__ x,
                                                      const int*   __restrict__ ei,
                                                      const int*   __restrict__ et,
                                                      const float* __restrict__ ew,
                                                      float* __restrict__ update) {
    const int e    = blockIdx.x * 8 + (threadIdx.x >> 5);
    const int lane = threadIdx.x & 31;
    const int src  = ei[e];
    const int dst  = ei[N_EDGES + e];
    const int ty   = et[e];
    const float w  = ew[e];
    const float2 v = ((const float2*)(x + (size_t)src * DIM))[lane];
    float* u = update + (((size_t)dst * NTYPE + ty) * DIM) + lane * 2;
    unsafeAtomicAdd(u,     w * v.x);
    unsafeAtomicAdd(u + 1, w * v.y);
}

// --------------------------- WMMA GEMM1: hidden = relu(update@WtM + b) -----
// Block = 8 waves, 16 nodes per block. Each wave: two 16-col tiles -> 256 cols.
// A tile staged in LDS (+4 pad -> conflict-free column reads); B read from the
// pre-transposed WtM[k][h] so lanes 0..15 hit contiguous floats.
__global__ __launch_bounds__(256) void gemm1_kernel(const float* __restrict__ update,
                                                    const float* __restrict__ WtM,
                                                    const float* __restrict__ bias,
                                                    float* __restrict__ hidden) {
    __shared__ float As[16][HID + 4];
    const int ntile = blockIdx.x;
    const float* Arow = update + (size_t)ntile * 16 * HID;
    for (int i = threadIdx.x; i < 16 * (HID / 4); i += 256) {
        int r = i / (HID / 4), c4 = i % (HID / 4);
        float4 v = ((const float4*)(Arow + (size_t)r * HID))[c4];
        *(float4*)&As[r][c4 * 4] = v;
    }
    __syncthreads();

    const int lane = threadIdx.x & 31;
    const int wave = threadIdx.x >> 5;
    const int nsel = lane & 15;            // A: M-row ; B/C: N-col
    const int koff = (lane >> 4) << 1;     // lanes 16-31 hold K+2,K+3
    const int mhi  = (lane >> 4) * 8;      // C: lanes 16-31 hold M=8..15

    for (int t = 0; t < 2; ++t) {
        const int hbase = wave * 32 + t * 16;
        v8f acc;
        const float b = bias[hbase + nsel];
#pragma unroll
        for (int i = 0; i < 8; ++i) acc[i] = b;

        for (int k0 = 0; k0 < HID; k0 += 4) {
            v2f a, bb;
            a.x = As[nsel][k0 + koff];
            a.y = As[nsel][k0 + koff + 1];
            const float* wp = WtM + (size_t)(k0 + koff) * HID + hbase + nsel;
            bb.x = wp[0];
            bb.y = wp[HID];
            acc = __builtin_amdgcn_wmma_f32_16x16x4_f32(false, a, false, bb,
                                                        (short)0, acc, false, false);
        }
        float* out = hidden + (size_t)ntile * 16 * HID + hbase + nsel;
#pragma unroll
        for (int r = 0; r < 8; ++r) {
            float v = acc[r];
            out[(size_t)(r + mhi) * HID] = v > 0.f ? v : 0.f;
        }
    }
}

// --------------------------- fused WMMA GEMM2/3 + GRU ----------------------
// 24 col-tiles of 16 (12 for gi over K=256, 12 for gh over K=64) spread over
// 8 waves (3 tiles each). Gate tiles land in LDS; elementwise GRU follows.
__global__ __launch_bounds__(256) void gru_kernel(const float* __restrict__ hidden,
                                                  const float* __restrict__ x,
                                                  const float* __restrict__ Wt_ih,
                                                  const float* __restrict__ Wt_hh,
                                                  const float* __restrict__ b_ih,
                                                  const float* __restrict__ b_hh,
                                                  float* __restrict__ out) {
    __shared__ float Hs[16][HID + 4];
    __shared__ float Xs[16][DIM + 4];
    __shared__ float Gi[16][G3 + 4];
    __shared__ float Gh[16][G3 + 4];

    const int ntile = blockIdx.x;
    const float* Hrow = hidden + (size_t)ntile * 16 * HID;
    const float* Xrow = x      + (size_t)ntile * 16 * DIM;
    for (int i = threadIdx.x; i < 16 * (HID / 4); i += 256) {
        int r = i / (HID / 4), c4 = i % (HID / 4);
        *(float4*)&Hs[r][c4 * 4] = ((const float4*)(Hrow + (size_t)r * HID))[c4];
    }
    {
        int i = threadIdx.x;            // 16 * (DIM/4) == 256 exactly
        int r = i / (DIM / 4), c4 = i % (DIM / 4);
        *(float4*)&Xs[r][c4 * 4] = ((const float4*)(Xrow + (size_t)r * DIM))[c4];
    }
    __syncthreads();

    const int lane = threadIdx.x & 31;
    const int wave = threadIdx.x >> 5;
    const int nsel = lane & 15;
    const int koff = (lane >> 4) << 1;
    const int mhi  = (lane >> 4) * 8;

    for (int i = 0; i < 3; ++i) {
        const int tt = wave * 3 + i;
        if (tt < 12) {                              // gi tile: K = 256 over Hs
            const int jb = tt * 16;
            v8f acc;
            const float b = b_ih[jb + nsel];
#pragma unroll
            for (int q = 0; q < 8; ++q) acc[q] = b;
            for (int k0 = 0; k0 < HID; k0 += 4) {
                v2f a, bb;
                a.x = Hs[nsel][k0 + koff];
                a.y = Hs[nsel][k0 + koff + 1];
                const float* wp = Wt_ih + (size_t)(k0 + koff) * G3 + jb + nsel;
                bb.x = wp[0];
                bb.y = wp[G3];
                acc = __builtin_amdgcn_wmma_f32_16x16x4_f32(false, a, false, bb,
                                                            (short)0, acc, false, false);
            }
#pragma unroll
            for (int r = 0; r < 8; ++r) Gi[r + mhi][jb + nsel] = acc[r];
        } else {                                    // gh tile: K = 64 over Xs
            const int jb = (tt - 12) * 16;
            v8f acc;
            const float b = b_hh[jb + nsel];
#pragma unroll
            for (int q = 0; q < 8; ++q) acc[q] = b;
            for (int k0 = 0; k0 < DIM; k0 += 4) {
                v2f a, bb;
                a.x = Xs[nsel][k0 + koff];
                a.y = Xs[nsel][k0 + koff + 1];
                const float* wp = Wt_hh + (size_t)(k0 + koff) * G3 + jb + nsel;
                bb.x = wp[0];
                bb.y = wp[G3];
                acc = __builtin_amdgcn_wmma_f32_16x16x4_f32(false, a, false, bb,
                                                            (short)0, acc, false, false);
            }
#pragma unroll
            for (int r = 0; r < 8; ++r) Gh[r + mhi][jb + nsel] = acc[r];
        }
    }
    __syncthreads();

    // GRU elementwise: 16 nodes x 64 dims = 1024 outputs, 4 per thread.
#pragma unroll
    for (int q = 0; q < 4; ++q) {
        const int idx = threadIdx.x + q * 256;
        const int r = idx >> 6, d = idx & 63;
        const float rr = 1.f / (1.f + __expf(-(Gi[r][d]       + Gh[r][d])));
        const float zz = 1.f / (1.f + __expf(-(Gi[r][64 + d]  + Gh[r][64 + d])));
        const float nn = tanhf(Gi[r][128 + d] + rr * Gh[r][128 + d]);
        const float h  = Xs[r][d];
        out[((size_t)ntile * 16 + r) * DIM + d] = (1.f - zz) * nn + zz * h;
    }
}

// ---------------------------------------------------------------------------
extern "C" void kernel_launch(void* const* d_in, const int* in_sizes, int n_in,
                              void* d_out, int out_size, void* d_ws, size_t ws_size,
                              hipStream_t stream) {
    const float* node_feature = (const float*)d_in[0];
    const int*   edge_index   = (const int*)  d_in[1];
    const int*   edge_type    = (const int*)  d_in[2];
    const float* edge_weight  = (const float*)d_in[3];
    const float* mlp_W        = (const float*)d_in[4];
    const float* mlp_b        = (const float*)d_in[5];
    const float* w_ih         = (const float*)d_in[6];
    const float* w_hh         = (const float*)d_in[7];
    const float* b_ih         = (const float*)d_in[8];
    const float* b_hh         = (const float*)d_in[9];
    float* out = (float*)d_out;

    // workspace layout (floats)
    float* ws     = (float*)d_ws;
    float* update = ws;                                        // N * 256
    float* hidden = update + (size_t)N_NODES * HID;            // N * 256
    float* WtM    = hidden + (size_t)N_NODES * HID;            // 256 * 256
    float* Wt_ih  = WtM    + (size_t)HID * HID;                // 256 * 192
    float* Wt_hh  = Wt_ih  + (size_t)HID * G3;                 // 64  * 192

    // 1) zero scatter buffer
    {
        int n4 = N_NODES * HID / 4;
        zero_kernel<<<(n4 + 255) / 256, 256, 0, stream>>>((float4*)update, n4);
    }
    // 2) pre-transpose weights -> [K, out] for coalesced WMMA B-fragment loads
    transpose_kernel<<<(HID * HID + 255) / 256, 256, 0, stream>>>(mlp_W, WtM,   HID, HID);
    transpose_kernel<<<(G3  * HID + 255) / 256, 256, 0, stream>>>(w_ih,  Wt_ih, G3,  HID);
    transpose_kernel<<<(G3  * DIM + 255) / 256, 256, 0, stream>>>(w_hh,  Wt_hh, G3,  DIM);
    // 3) edge scatter (one wave32 per edge)
    scatter_kernel<<<N_EDGES / 8, 256, 0, stream>>>(node_feature, edge_index,
                                                    edge_type, edge_weight, update);
    // 4) hidden = relu(update @ mlp_W^T + mlp_b)   (fp32 WMMA)
    gemm1_kernel<<<N_NODES / 16, 256, 0, stream>>>(update, WtM, mlp_b, hidden);
    // 5) fused gates + GRU                          (fp32 WMMA)
    gru_kernel<<<N_NODES / 16, 256, 0, stream>>>(hidden, node_feature,
                                                 Wt_ih, Wt_hh, b_ih, b_hh, out);
}